// GraphBasedEmbedder_4063039062508
// MI455X (gfx1250) — compile-verified
//
#include <hip/hip_runtime.h>

typedef float v2f __attribute__((ext_vector_type(2)));
typedef float v8f __attribute__((ext_vector_type(8)));

// ---------------------------------------------------------------------------
// k1: deg[i] = 1.0 (self-loop contribution folded into init)
// ---------------------------------------------------------------------------
__global__ void k_init_deg(float* __restrict__ deg, int n) {
    int i = blockIdx.x * blockDim.x + threadIdx.x;
    if (i < n) deg[i] = 1.0f;
}

// ---------------------------------------------------------------------------
// k2: deg[col[e]] += 1 over all edges
// ---------------------------------------------------------------------------
__global__ void k_edge_deg(const int* __restrict__ col, float* __restrict__ deg, int e) {
    int i = blockIdx.x * blockDim.x + threadIdx.x;
    if (i < e) atomicAdd(&deg[col[i]], 1.0f);
}

// ---------------------------------------------------------------------------
// k3: dinv[i] = rsqrt(deg[i])   (deg >= 1 always, self-loop guaranteed)
// ---------------------------------------------------------------------------
__global__ void k_dinv(const float* __restrict__ deg, float* __restrict__ dinv, int n) {
    int i = blockIdx.x * blockDim.x + threadIdx.x;
    if (i < n) {
        float d = deg[i];
        dinv[i] = (d > 0.0f) ? rsqrtf(d) : 0.0f;
    }
}

// ---------------------------------------------------------------------------
// k4: X = A(MxK) * B(KxN)  via V_WMMA_F32_16X16X4_F32, K=N=128.
// One wave per 16x16 output tile; 8 waves (256 threads) per block.
// Fragment layouts per CDNA5 ISA 7.12.2:
//   A 16x4 (f32): lane L, m = L%16; lanes 0-15 hold K=k0,k0+1 in v0,v1;
//                 lanes 16-31 hold K=k0+2,k0+3.
//   B 4x16 (f32): row K striped across lanes; lanes 0-15 rows k0,k0+1;
//                 lanes 16-31 rows k0+2,k0+3. n = L%16.
//   C/D 16x16:    VGPR j, lane L -> M = j + 8*(L/16), N = L%16.
// ---------------------------------------------------------------------------
__global__ void __launch_bounds__(256) k_gemm_wmma(const float* __restrict__ A,
                                                  const float* __restrict__ B,
                                                  float* __restrict__ X,
                                                  int M) {
    const int K  = 128;
    const int NC = 128;
    int wave = threadIdx.x >> 5;
    int lane = threadIdx.x & 31;
    int gw   = blockIdx.x * 8 + wave;

    int row_tile = gw >> 3;   // M/16 tiles
    int col_tile = gw & 7;    // 128/16 = 8 column tiles
    int half  = lane >> 4;    // 0 or 1
    int idx16 = lane & 15;

    int row = row_tile * 16 + idx16;
    int rr  = (row < M) ? row : (M - 1);     // clamp (M is a multiple of 16 here)
    int ncol = col_tile * 16 + idx16;

    const float* arow = A + (size_t)rr * K;

    v8f c = {};
    for (int k0 = 0; k0 < K; k0 += 4) {
        int ka = k0 + 2 * half;
        v2f a = *(const v2f*)(arow + ka);            // A[m][ka], A[m][ka+1]
        v2f b;
        b.x = B[(size_t)ka * NC + ncol];             // W[ka][n]
        b.y = B[(size_t)(ka + 1) * NC + ncol];       // W[ka+1][n]
        c = __builtin_amdgcn_wmma_f32_16x16x4_f32(
                /*neg_a=*/false, a, /*neg_b=*/false, b,
                /*c_mod=*/(short)0, c, /*reuse_a=*/false, /*reuse_b=*/false);
    }

#pragma unroll
    for (int j = 0; j < 8; ++j) {
        int orow = row_tile * 16 + j + 8 * half;
        if (orow < M) X[(size_t)orow * NC + ncol] = c[j];
    }
}

// ---------------------------------------------------------------------------
// k5: out[i][j] = bias[j] + dinv[i]^2 * x[i][j]   (self-loop message + bias)
// Non-atomic: runs before the edge-scatter kernel on the same stream.
// ---------------------------------------------------------------------------
__global__ void k_self_bias(const float* __restrict__ x,
                            const float* __restrict__ dinv,
                            const float* __restrict__ bias,
                            float* __restrict__ out, int n) {
    int idx = blockIdx.x * blockDim.x + threadIdx.x;
    int total = n * 128;
    if (idx < total) {
        int i = idx >> 7;
        int j = idx & 127;
        float di = dinv[i];
        out[idx] = bias[j] + di * di * x[idx];
    }
}

// ---------------------------------------------------------------------------
// k6: edge scatter. One wave32 per edge: each lane handles 4 features
// (128 = 32 lanes x float4). x and out are L2-resident (51.2 MB each vs
// 192 MB L2), so the random gather + atomic scatter stay on-chip.
// ---------------------------------------------------------------------------
__global__ void __launch_bounds__(256) k_scatter(const int* __restrict__ row,
                                                 const int* __restrict__ col,
                                                 const float* __restrict__ dinv,
                                                 const float* __restrict__ x,
                                                 float* __restrict__ out, int e) {
    int tid  = blockIdx.x * blockDim.x + threadIdx.x;
    int wave = tid >> 5;
    int lane = tid & 31;
    if (wave >= e) return;

    int r = row[wave];
    int c = col[wave];
    float nrm = dinv[r] * dinv[c];

    const float4* xr = (const float4*)(x + (size_t)r * 128);
    float4 v = xr[lane];                       // global_load_b128

    float* op = out + (size_t)c * 128 + lane * 4;
    atomicAdd(op + 0, nrm * v.x);
    atomicAdd(op + 1, nrm * v.y);
    atomicAdd(op + 2, nrm * v.z);
    atomicAdd(op + 3, nrm * v.w);
}

// ---------------------------------------------------------------------------
extern "C" void kernel_launch(void* const* d_in, const int* in_sizes, int n_in,
                              void* d_out, int out_size, void* d_ws, size_t ws_size,
                              hipStream_t stream) {
    const float* features = (const float*)d_in[0];
    const int*   edge_idx = (const int*)d_in[1];   // [2, E] flat: row then col
    // d_in[2] = batch (unused by the reference output)
    const float* W    = (const float*)d_in[3];
    const float* bias = (const float*)d_in[4];

    const int N = in_sizes[0] / 128;
    const int E = in_sizes[1] / 2;
    const int* row = edge_idx;
    const int* col = edge_idx + E;

    float* out  = (float*)d_out;
    float* x    = (float*)d_ws;                 // N*128 floats (51.2 MB)
    float* deg  = x + (size_t)N * 128;          // N floats
    float* dinv = deg + N;                      // N floats

    const int TB = 256;
    int nb_n  = (N + TB - 1) / TB;
    int nb_e  = (E + TB - 1) / TB;
    int m16   = (N + 15) / 16;
    int nb_sb = ((size_t)N * 128 + TB - 1) / TB;
    int nb_sc = (E + 7) / 8;                    // 8 waves per block, 1 wave/edge

    k_init_deg <<<nb_n,  TB, 0, stream>>>(deg, N);
    k_edge_deg <<<nb_e,  TB, 0, stream>>>(col, deg, E);
    k_dinv     <<<nb_n,  TB, 0, stream>>>(deg, dinv, N);
    k_gemm_wmma<<<m16,   TB, 0, stream>>>(features, W, x, N);
    k_self_bias<<<nb_sb, TB, 0, stream>>>(x, dinv, bias, out, N);
    k_scatter  <<<nb_sc, TB, 0, stream>>>(row, col, dinv, x, out, E);
}